// PANPooling_128849019307
// MI455X (gfx1250) — compile-verified
//
#include <hip/hip_runtime.h>
#include <hip/hip_bf16.h>
#include <math.h>

#define NN   131072
#define BB   64
#define CC   128
#define EE   4194304
#define NPER 2048
#define KPER 1024
#define KK   65536

#define SCAN_THREADS 256
#define SCAN_ITEMS   16
#define SCAN_CHUNK   (SCAN_THREADS * SCAN_ITEMS)   // 4096
#define SCAN_NBLK    (EE / SCAN_CHUNK)             // 1024

typedef float v2f __attribute__((ext_vector_type(2)));
typedef float v8f __attribute__((ext_vector_type(8)));

// ---------------- init: degree = 0, nodemap = -1 ----------------
__global__ void k_init(float* __restrict__ degree, int* __restrict__ nodemap) {
    int n = blockIdx.x * blockDim.x + threadIdx.x;
    if (n < NN) { degree[n] = 0.0f; nodemap[n] = -1; }
}

// ---------------- degree = segment_sum(edge_value, row) ----------------
__global__ void k_degree(const int* __restrict__ ei, const float* __restrict__ ew,
                         float* __restrict__ degree) {
    int e = blockIdx.x * blockDim.x + threadIdx.x;
    if (e < EE) atomicAdd(&degree[ei[e]], ew[e]);
}

// ---------------- score = tanh(w0 * (x @ transform) + w1 * degree) ----------------
// One wave computes 16 node dot-products via V_WMMA_F32_16X16X4_F32,
// accumulating 32 chunks of K=4. transform occupies column 0 of B, so
// column 0 of the 16x16 accumulator is the matvec result.
// A 16x4 f32 layout: lanes 0-15 -> {K0,K1}, lanes 16-31 -> {K2,K3}, M = lane&15.
// B 4x16 f32 layout (mirror): lane 0 carries {K0,K1} of column 0, lane 16 {K2,K3}.
// B fragment is built BRANCH-FREE: every lane loads its float2 of transform
// (lo lanes {t[kb],t[kb+1]}, hi lanes {t[kb+2],t[kb+3]}) and v_cndmask's it
// to zero unless m==0, avoiding exec save/restore in the WMMA loop.
// D layout: N = lane (0-15) / lane-16 (16-31); VGPR v holds M=v (lo), M=v+8 (hi).
__global__ void k_score(const float* __restrict__ x, const float* __restrict__ transform,
                        const float* __restrict__ w, const float* __restrict__ degree,
                        float* __restrict__ score) {
    const int lane     = threadIdx.x & 31;
    const int wave     = threadIdx.x >> 5;
    const int waveId   = blockIdx.x * (blockDim.x >> 5) + wave;  // grid tiles N exactly
    const int nodeBase = waveId * 16;
    const int m        = lane & 15;
    const int khalf    = (lane >> 4) << 1;                       // 0 or 2
    const bool bsel    = (m == 0);                               // lanes 0 and 16 carry B col 0

    const float* xrow = x + (size_t)(nodeBase + m) * CC + khalf;
    const float* trow = transform + khalf;
    v8f acc = {};
    #pragma unroll 4
    for (int kb = 0; kb < CC; kb += 4) {
        v2f a;
        a.x = xrow[kb];
        a.y = xrow[kb + 1];
        float tx = trow[kb];
        float ty = trow[kb + 1];
        v2f b;
        b.x = bsel ? tx : 0.0f;
        b.y = bsel ? ty : 0.0f;
        acc = __builtin_amdgcn_wmma_f32_16x16x4_f32(
            /*neg_a=*/false, a, /*neg_b=*/false, b,
            /*c_mod=*/(short)0, acc, /*reuse_a=*/false, /*reuse_b=*/false);
    }
    const float w0 = w[0], w1 = w[1];
    if (lane == 0 || lane == 16) {
        const int base = nodeBase + (lane == 16 ? 8 : 0);
        #pragma unroll
        for (int r = 0; r < 8; ++r) {
            const int n = base + r;
            score[n] = tanhf(w0 * acc[r] + w1 * degree[n]);
        }
    }
}

// ---------------- per-graph stable descending top-k via LDS bitonic sort ----------------
__device__ __forceinline__ bool before(float ka, int ia, float kb, int ib) {
    // descending score, stable (smaller original index first on ties)
    return (ka > kb) || (ka == kb && ia < ib);
}

__global__ void __launch_bounds__(1024)
k_sort(const float* __restrict__ score, int* __restrict__ perm_ws,
       int* __restrict__ nodemap, int* __restrict__ perm_out) {
    __shared__ float skey[NPER];
    __shared__ int   sidx[NPER];
    const int g = blockIdx.x;
    const int t = threadIdx.x;

    for (int i = t; i < NPER; i += blockDim.x) {
        skey[i] = score[g * NPER + i];
        sidx[i] = i;
    }
    for (int k = 2; k <= NPER; k <<= 1) {
        for (int j = k >> 1; j > 0; j >>= 1) {
            __syncthreads();
            for (int i = t; i < NPER; i += blockDim.x) {
                const int ixj = i ^ j;
                if (ixj > i) {
                    const bool dir = ((i & k) == 0);
                    const float ka = skey[i],  kb = skey[ixj];
                    const int   ia = sidx[i],  ib = sidx[ixj];
                    const bool b_first = before(kb, ib, ka, ia);
                    if (b_first == dir) {   // enforce segment direction
                        skey[i] = kb; skey[ixj] = ka;
                        sidx[i] = ib; sidx[ixj] = ia;
                    }
                }
            }
        }
    }
    __syncthreads();
    for (int i = t; i < KPER; i += blockDim.x) {
        const int gid  = g * NPER + sidx[i];
        const int outi = g * KPER + i;
        perm_ws[outi]  = gid;
        perm_out[outi] = gid;
        nodemap[gid]   = outi;
    }
}

// ---------------- batch_out / score_perm ----------------
__global__ void k_small(const int* __restrict__ batch, const float* __restrict__ score,
                        const int* __restrict__ perm_ws,
                        int* __restrict__ batch_out, float* __restrict__ sperm) {
    int i = blockIdx.x * blockDim.x + threadIdx.x;
    if (i < KK) {
        const int p = perm_ws[i];
        batch_out[i] = batch[p];
        sperm[i]     = score[p];
    }
}

// ---------------- x_out = x[perm] * score_perm[:,None], float4 ----------------
__global__ void k_gather(const float* __restrict__ x, const float* __restrict__ score,
                         const int* __restrict__ perm_ws, float* __restrict__ x_out) {
    const int tid = blockIdx.x * blockDim.x + threadIdx.x;   // KK * (CC/4) threads
    const int i   = tid >> 5;
    const int c4  = (tid & 31) << 2;
    const int p   = perm_ws[i];
    const float s = score[p];
    const float4 v = *(const float4*)(x + (size_t)p * CC + c4);
    float4 o;
    o.x = v.x * s; o.y = v.y * s; o.z = v.z * s; o.w = v.w * s;
    *(float4*)(x_out + (size_t)i * CC + c4) = o;
}

// ---------------- edge outputs default fill: (-1, -1, 0.0) ----------------
__global__ void k_fill(int* __restrict__ ro, int* __restrict__ co, float* __restrict__ eo) {
    int e = blockIdx.x * blockDim.x + threadIdx.x;
    if (e < EE) { ro[e] = -1; co[e] = -1; eo[e] = 0.0f; }
}

// ---------------- scan phase 1: per-block valid-edge counts ----------------
__global__ void k_scan1(const int* __restrict__ ei, const int* __restrict__ nodemap,
                        unsigned* __restrict__ counts) {
    __shared__ unsigned sh[SCAN_THREADS];
    const int b = blockIdx.x, t = threadIdx.x;
    const int base = b * SCAN_CHUNK + t * SCAN_ITEMS;
    unsigned cnt = 0;
    #pragma unroll
    for (int j = 0; j < SCAN_ITEMS; ++j) {
        const int e = base + j;
        cnt += (nodemap[ei[e]] >= 0) && (nodemap[ei[EE + e]] >= 0);
    }
    sh[t] = cnt; __syncthreads();
    for (int off = SCAN_THREADS >> 1; off > 0; off >>= 1) {
        if (t < off) sh[t] += sh[t + off];
        __syncthreads();
    }
    if (t == 0) counts[b] = sh[0];
}

// ---------------- scan phase 2: exclusive scan of block counts (single block) --------
__global__ void __launch_bounds__(SCAN_NBLK)
k_scan2(const unsigned* __restrict__ counts, unsigned* __restrict__ offsets) {
    __shared__ unsigned sh[SCAN_NBLK];
    const int t = threadIdx.x;
    const unsigned v = counts[t];
    sh[t] = v; __syncthreads();
    for (int off = 1; off < SCAN_NBLK; off <<= 1) {
        const unsigned add = (t >= off) ? sh[t - off] : 0u;
        __syncthreads();
        sh[t] += add;
        __syncthreads();
    }
    offsets[t] = sh[t] - v;   // exclusive
}

// ---------------- scan phase 3: stable scatter of valid edges ----------------
__global__ void k_scan3(const int* __restrict__ ei, const float* __restrict__ ew,
                        const int* __restrict__ nodemap, const unsigned* __restrict__ offsets,
                        int* __restrict__ ro, int* __restrict__ co, float* __restrict__ eo) {
    __shared__ unsigned sh[SCAN_THREADS];
    const int b = blockIdx.x, t = threadIdx.x;
    const int base = b * SCAN_CHUNK + t * SCAN_ITEMS;
    int rr[SCAN_ITEMS], cc[SCAN_ITEMS];
    unsigned fl = 0, cnt = 0;
    #pragma unroll
    for (int j = 0; j < SCAN_ITEMS; ++j) {
        const int e = base + j;
        const int r = nodemap[ei[e]];
        const int c = nodemap[ei[EE + e]];
        rr[j] = r; cc[j] = c;
        const bool v = (r >= 0) && (c >= 0);
        fl |= (unsigned)v << j;
        cnt += v;
    }
    sh[t] = cnt; __syncthreads();
    for (int off = 1; off < SCAN_THREADS; off <<= 1) {
        const unsigned add = (t >= off) ? sh[t - off] : 0u;
        __syncthreads();
        sh[t] += add;
        __syncthreads();
    }
    unsigned pos = offsets[b] + sh[t] - cnt;   // block offset + in-block exclusive rank
    #pragma unroll
    for (int j = 0; j < SCAN_ITEMS; ++j) {
        if ((fl >> j) & 1u) {
            ro[pos] = rr[j];
            co[pos] = cc[j];
            eo[pos] = ew[base + j];
            ++pos;
        }
    }
}

extern "C" void kernel_launch(void* const* d_in, const int* in_sizes, int n_in,
                              void* d_out, int out_size, void* d_ws, size_t ws_size,
                              hipStream_t stream) {
    const float* x   = (const float*)d_in[0];   // [N, C]
    const float* ev  = (const float*)d_in[1];   // [E]
    const float* tr  = (const float*)d_in[2];   // [C]
    const float* w   = (const float*)d_in[3];   // [2]
    const int*   ei  = (const int*)d_in[4];     // [2, E]
    const int*   bat = (const int*)d_in[5];     // [N]

    // workspace carve (~1.84 MB)
    float*    degree  = (float*)d_ws;
    float*    score   = degree + NN;
    int*      perm_ws = (int*)(score + NN);
    int*      nodemap = perm_ws + KK;
    unsigned* counts  = (unsigned*)(nodemap + NN);
    unsigned* offsets = counts + SCAN_NBLK;

    // output carve (flat, return order); int outputs bit-stored via aliases
    float* out        = (float*)d_out;
    float* x_out      = out;                               // K*C
    int*   row_out    = (int*)(out + (size_t)KK * CC);     // E
    int*   col_out    = row_out + EE;                      // E
    float* ew_out     = (float*)(col_out + EE);            // E
    int*   batch_out  = (int*)(ew_out + EE);               // K
    int*   perm_out   = batch_out + KK;                    // K
    float* sperm_out  = (float*)(perm_out + KK);           // K

    hipLaunchKernelGGL(k_init,   dim3(NN / 256),        dim3(256),          0, stream, degree, nodemap);
    hipLaunchKernelGGL(k_degree, dim3(EE / 256),        dim3(256),          0, stream, ei, ev, degree);
    hipLaunchKernelGGL(k_score,  dim3(NN / 128),        dim3(256),          0, stream, x, tr, w, degree, score);
    hipLaunchKernelGGL(k_sort,   dim3(BB),              dim3(1024),         0, stream, score, perm_ws, nodemap, perm_out);
    hipLaunchKernelGGL(k_small,  dim3(KK / 256),        dim3(256),          0, stream, bat, score, perm_ws, batch_out, sperm_out);
    hipLaunchKernelGGL(k_gather, dim3((KK * 32) / 256), dim3(256),          0, stream, x, score, perm_ws, x_out);
    hipLaunchKernelGGL(k_fill,   dim3(EE / 256),        dim3(256),          0, stream, row_out, col_out, ew_out);
    hipLaunchKernelGGL(k_scan1,  dim3(SCAN_NBLK),       dim3(SCAN_THREADS), 0, stream, ei, nodemap, counts);
    hipLaunchKernelGGL(k_scan2,  dim3(1),               dim3(SCAN_NBLK),    0, stream, counts, offsets);
    hipLaunchKernelGGL(k_scan3,  dim3(SCAN_NBLK),       dim3(SCAN_THREADS), 0, stream, ei, ev, nodemap, offsets, row_out, col_out, ew_out);
}